// HistogramClamp_64415919506098
// MI455X (gfx1250) — compile-verified
//
#include <hip/hip_runtime.h>
#include <cmath>
#include <cstdint>

// ---------------------------------------------------------------------------
// Exact radix-select quantile (k-th order statistic of |x|) + clamp.
//   ws layout (uint32): [0, 32768)           hist_hi (top 15 bits of |x| bits)
//                       [32768, 32768+65536) hist_lo (low 16 bits, selected bin)
//                       [98304, 98312)       sel: {hi_bin, residual_rank, cv_bits, ...}
// ---------------------------------------------------------------------------

#ifndef __has_builtin
#define __has_builtin(x) 0
#endif

#if __has_builtin(__builtin_amdgcn_global_load_async_to_lds_b128) && \
    __has_builtin(__builtin_amdgcn_s_wait_asynccnt)
#define USE_ASYNC_LDS 1
#else
#define USE_ASYNC_LDS 0
#endif

#define HI_BINS 32768
#define LO_BINS 65536
#define HB_THREADS 1024

#if USE_ASYNC_LDS
typedef int v4i_t __attribute__((vector_size(16)));
// Builtin prototype (from hipcc diagnostic): (v4i as1*, v4i as3*, imm i32, imm i32)
#define GLB_V4(p) ((__attribute__((address_space(1))) v4i_t*)(void*)(p))
#define LDS_V4(p) ((__attribute__((address_space(3))) v4i_t*)(void*)(p))
#endif

__global__ void zero_ws_kernel(unsigned* __restrict__ w, int n) {
  int i = blockIdx.x * blockDim.x + threadIdx.x;
  int s = gridDim.x * blockDim.x;
  for (; i < n; i += s) w[i] = 0u;
}

// ---------------------------------------------------------------------------
// Pass 1: LDS-privatized histogram over bits(|x|) >> 16  (32768 bins, 128KB LDS)
// Streams x through LDS with double-buffered async global->LDS loads.
// ---------------------------------------------------------------------------
__global__ __launch_bounds__(HB_THREADS)
void hist_hi_kernel(const float* __restrict__ x, long long n,
                    unsigned* __restrict__ ghist) {
  __shared__ unsigned lh[HI_BINS];
  for (int i = threadIdx.x; i < HI_BINS; i += HB_THREADS) lh[i] = 0u;
  __syncthreads();

  auto bump4 = [&](float4 v) {
    atomicAdd(&lh[(__float_as_uint(v.x) & 0x7FFFFFFFu) >> 16], 1u);
    atomicAdd(&lh[(__float_as_uint(v.y) & 0x7FFFFFFFu) >> 16], 1u);
    atomicAdd(&lh[(__float_as_uint(v.z) & 0x7FFFFFFFu) >> 16], 1u);
    atomicAdd(&lh[(__float_as_uint(v.w) & 0x7FFFFFFFu) >> 16], 1u);
  };

  const long long n4 = n >> 2;
  const float4* x4 = (const float4*)x;
  const long long ntiles = n4 / HB_THREADS;  // full tiles of 1024 float4

#if USE_ASYNC_LDS
  __shared__ float4 stage[2][HB_THREADS];
  long long t = blockIdx.x;
  if (t < ntiles) {
    const float4* g = x4 + t * HB_THREADS + threadIdx.x;
    __builtin_amdgcn_global_load_async_to_lds_b128(
        GLB_V4(g), LDS_V4(&stage[0][threadIdx.x]), 0, 0);
  }
  int buf = 0;
  for (; t < ntiles; t += gridDim.x) {
    long long tn = t + gridDim.x;
    if (tn < ntiles) {
      const float4* g = x4 + tn * HB_THREADS + threadIdx.x;
      __builtin_amdgcn_global_load_async_to_lds_b128(
          GLB_V4(g), LDS_V4(&stage[buf ^ 1][threadIdx.x]), 0, 0);
      __builtin_amdgcn_s_wait_asynccnt(1);  // oldest (current buf) complete
    } else {
      __builtin_amdgcn_s_wait_asynccnt(0);
    }
    bump4(stage[buf][threadIdx.x]);  // ds_load_b128 + ds_add_u32 x4
    buf ^= 1;
  }
#else
  {
    long long stride = (long long)gridDim.x * HB_THREADS;
    for (long long i = (long long)blockIdx.x * HB_THREADS + threadIdx.x;
         i < ntiles * HB_THREADS; i += stride)
      bump4(x4[i]);
  }
#endif

  // remainder float4s beyond full tiles
  {
    long long stride = (long long)gridDim.x * HB_THREADS;
    for (long long i = ntiles * HB_THREADS + (long long)blockIdx.x * HB_THREADS + threadIdx.x;
         i < n4; i += stride)
      bump4(x4[i]);
    // scalar remainder
    for (long long i = (n4 << 2) + (long long)blockIdx.x * HB_THREADS + threadIdx.x;
         i < n; i += stride)
      atomicAdd(&lh[(__float_as_uint(x[i]) & 0x7FFFFFFFu) >> 16], 1u);
  }

  __syncthreads();
  for (int i = threadIdx.x; i < HI_BINS; i += HB_THREADS) {
    unsigned c = lh[i];
    if (c) atomicAdd(&ghist[i], c);
  }
}

// ---------------------------------------------------------------------------
// Single-workgroup scan over a histogram; finds the bin containing `rank`.
// stage 0: hist_hi -> sel[0]=hi_bin, sel[1]=residual rank
// stage 1: hist_lo -> sel[2]=cv_bits = (hi_bin<<16)|lo_bin
// ---------------------------------------------------------------------------
__global__ __launch_bounds__(1024)
void radix_select_step(const unsigned* __restrict__ hist, int nbins,
                       unsigned* __restrict__ sel, unsigned k, int stage) {
  __shared__ unsigned part[1024];
  const int tid = threadIdx.x;
  const int per = nbins / 1024;
  const unsigned rank = (stage == 0) ? k : sel[1];
  const int base = tid * per;

  unsigned s = 0;
  for (int i = 0; i < per; ++i) s += hist[base + i];
  part[tid] = s;
  __syncthreads();
  for (int off = 1; off < 1024; off <<= 1) {  // inclusive Hillis-Steele scan
    unsigned v = (tid >= off) ? part[tid - off] : 0u;
    __syncthreads();
    part[tid] += v;
    __syncthreads();
  }
  const unsigned incl = part[tid];
  const unsigned excl = incl - s;
  if (rank >= excl && rank < incl) {
    unsigned c = excl;
    for (int i = 0; i < per; ++i) {
      unsigned h = hist[base + i];
      if (rank < c + h) {
        if (stage == 0) {
          sel[0] = (unsigned)(base + i);
          sel[1] = rank - c;
        } else {
          sel[2] = (sel[0] << 16) | (unsigned)(base + i);
        }
        break;
      }
      c += h;
    }
  }
}

// ---------------------------------------------------------------------------
// Pass 2: low-16-bit histogram, only for elements in the selected high bin.
// Sparse -> direct global atomics (expected ~n/32768 hits).
// ---------------------------------------------------------------------------
__global__ void hist_lo_kernel(const float* __restrict__ x, long long n,
                               const unsigned* __restrict__ sel,
                               unsigned* __restrict__ glo) {
  const unsigned b = sel[0];
  const long long gid = (long long)blockIdx.x * blockDim.x + threadIdx.x;
  const long long stride = (long long)gridDim.x * blockDim.x;
  const long long n4 = n >> 2;
  const float4* x4 = (const float4*)x;

  auto test = [&](float f) {
    unsigned kb = __float_as_uint(f) & 0x7FFFFFFFu;
    if ((kb >> 16) == b) atomicAdd(&glo[kb & 0xFFFFu], 1u);
  };
  for (long long i = gid; i < n4; i += stride) {
    float4 v = x4[i];
    test(v.x); test(v.y); test(v.z); test(v.w);
  }
  for (long long i = (n4 << 2) + gid; i < n; i += stride) test(x[i]);
}

// ---------------------------------------------------------------------------
// Pass 3: out = clip(x, -cv, cv), vectorized b128 streaming.
// ---------------------------------------------------------------------------
__global__ void clamp_kernel(const float* __restrict__ x, float* __restrict__ out,
                             long long n, const unsigned* __restrict__ sel) {
  const float cv = __uint_as_float(sel[2]);
  const long long gid = (long long)blockIdx.x * blockDim.x + threadIdx.x;
  const long long stride = (long long)gridDim.x * blockDim.x;
  const long long n4 = n >> 2;
  const float4* x4 = (const float4*)x;
  float4* o4 = (float4*)out;

  for (long long i = gid; i < n4; i += stride) {
    float4 v = x4[i];
    float4 r;
    r.x = fminf(fmaxf(v.x, -cv), cv);
    r.y = fminf(fmaxf(v.y, -cv), cv);
    r.z = fminf(fmaxf(v.z, -cv), cv);
    r.w = fminf(fmaxf(v.w, -cv), cv);
    o4[i] = r;
  }
  for (long long i = (n4 << 2) + gid; i < n; i += stride)
    out[i] = fminf(fmaxf(x[i], -cv), cv);
}

// ---------------------------------------------------------------------------
extern "C" void kernel_launch(void* const* d_in, const int* in_sizes, int n_in,
                              void* d_out, int out_size, void* d_ws, size_t ws_size,
                              hipStream_t stream) {
  (void)n_in; (void)out_size; (void)ws_size;
  const float* x = (const float*)d_in[0];  // d_in[1] = clamp_value buffer (unused for output)
  const long long n = (long long)in_sizes[0];

  unsigned* ws = (unsigned*)d_ws;
  unsigned* hist_hi = ws;
  unsigned* hist_lo = ws + HI_BINS;
  unsigned* sel = ws + HI_BINS + LO_BINS;

  // k = round(0.99 * n) - 1, matching the reference's static order-statistic index
  const double p = 99.0 / 100.0;
  long long kk = (long long)llround(p * (double)n) - 1;
  if (kk < 0) kk = 0;
  if (kk >= n) kk = n - 1;
  const unsigned k = (unsigned)kk;

  zero_ws_kernel<<<128, 256, 0, stream>>>(ws, HI_BINS + LO_BINS + 8);
  hist_hi_kernel<<<1024, HB_THREADS, 0, stream>>>(x, n, hist_hi);
  radix_select_step<<<1, 1024, 0, stream>>>(hist_hi, HI_BINS, sel, k, 0);
  hist_lo_kernel<<<4096, 256, 0, stream>>>(x, n, sel, hist_lo);
  radix_select_step<<<1, 1024, 0, stream>>>(hist_lo, LO_BINS, sel, k, 1);
  clamp_kernel<<<4096, 256, 0, stream>>>(x, (float*)d_out, n, sel);
}